// NumDecoder_55877524521068
// MI455X (gfx1250) — compile-verified
//
#include <hip/hip_runtime.h>
#include <hip/hip_bf16.h>
#include <stdint.h>

#define B_ 32
#define T_ 64
#define S_ 64
#define H_ 512
#define E_ 512
#define V_ 32000
#define H3_ 1536
#define ROWS_ (T_*B_)   // 2048

typedef __attribute__((ext_vector_type(2))) float v2f;
typedef __attribute__((ext_vector_type(8))) float v8f;
typedef __attribute__((ext_vector_type(4))) unsigned int u32x4;
typedef __attribute__((ext_vector_type(8))) int i32x8;
typedef __attribute__((ext_vector_type(4))) int i32x4;

#if defined(__has_builtin)
#if __has_builtin(__builtin_amdgcn_tensor_load_to_lds)
#define HAVE_TDM 1
#endif
#endif

// 16x16 fp32 tile GEMM over K using V_WMMA_F32_16X16X4_F32.
// aPtr: per-lane &A[m][2*hi]   (A row-major, row m = lane&15)
// bPtr: per-lane &W[n][2*hi]   (W row-major [N][K], n = lane&15; B[k][n] = W[n][k])
__device__ __forceinline__ v8f wmma_f32_k(const float* __restrict__ aPtr,
                                          const float* __restrict__ bPtr,
                                          int K, v8f acc) {
#pragma unroll 8
  for (int k = 0; k < K; k += 4) {
    v2f a = *(const v2f*)(aPtr + k);
    v2f b = *(const v2f*)(bPtr + k);
    acc = __builtin_amdgcn_wmma_f32_16x16x4_f32(false, a, false, b,
                                                (short)0, acc, false, false);
  }
  return acc;
}

__device__ __forceinline__ float sigm(float x) { return 1.0f / (1.0f + expf(-x)); }

// h0/h1 <- encoder_hidden[0], encoder_hidden[1]
__global__ void k_init(const float* __restrict__ eh, float* __restrict__ h0,
                       float* __restrict__ h1) {
  int i = blockIdx.x * blockDim.x + threadIdx.x;
  if (i < B_ * H_) {
    h0[i] = eh[i];
    h1[i] = eh[B_ * H_ + i];
  }
}

// gi0[t*B+b][:] = emb[x_{t,b}] @ W_ih0^T + b_ih0   (parallel over all t)
__global__ void k_pre(const int* __restrict__ target, const float* __restrict__ emb,
                      const float* __restrict__ W_ih0, const float* __restrict__ b_ih0,
                      float* __restrict__ gi0) {
  int lane = threadIdx.x & 31;
  int wave = threadIdx.x >> 5;
  int tile = blockIdx.x * 8 + wave;      // 128 row-blocks x 96 col-blocks
  int rb = tile / 96, cb = tile % 96;
  int m = lane & 15, hi = lane >> 4;
  int row = rb * 16 + m;                  // row = t*B + b
  int t = row >> 5, b = row & 31;
  int x = (t == 0) ? 1 : target[b * T_ + (t - 1)];  // SOS=1 teacher forcing
  const float* aPtr = emb + (size_t)x * E_ + 2 * hi;
  int col = cb * 16 + m;
  const float* bPtr = W_ih0 + (size_t)col * E_ + 2 * hi;
  v8f acc = {};
  acc = wmma_f32_k(aPtr, bPtr, E_, acc);
  float bias = b_ih0[col];
#pragma unroll
  for (int r = 0; r < 8; ++r) {
    int rc = rb * 16 + r + 8 * hi;
    gi0[(size_t)rc * H3_ + col] = acc[r] + bias;
  }
}

// GRU layer 0: gh0 = h0 @ W_hh0^T + b_hh0; combine with precomputed gi0 -> h0'
__global__ void k_gates0(int t, const float* __restrict__ gi0,
                         const float* __restrict__ W_hh0, const float* __restrict__ b_hh0,
                         const float* __restrict__ h0old, float* __restrict__ h0new) {
  int lane = threadIdx.x & 31;
  int rb = threadIdx.x >> 5;   // 2 waves: rows 0-15, 16-31
  int cb = blockIdx.x;         // 32 col-blocks of H
  int m = lane & 15, hi = lane >> 4;
  const float* aPtr = h0old + (size_t)(rb * 16 + m) * H_ + 2 * hi;
  int col = cb * 16 + m;
  v8f ar = {}, az = {}, an = {};
  ar = wmma_f32_k(aPtr, W_hh0 + (size_t)(0 * H_ + col) * H_ + 2 * hi, H_, ar);
  az = wmma_f32_k(aPtr, W_hh0 + (size_t)(1 * H_ + col) * H_ + 2 * hi, H_, az);
  an = wmma_f32_k(aPtr, W_hh0 + (size_t)(2 * H_ + col) * H_ + 2 * hi, H_, an);
  float br = b_hh0[col], bz = b_hh0[H_ + col], bn = b_hh0[2 * H_ + col];
  const float* gi = gi0 + (size_t)t * B_ * H3_;
#pragma unroll
  for (int r = 0; r < 8; ++r) {
    int rc = rb * 16 + r + 8 * hi;   // batch row 0..31
    float gr = gi[(size_t)rc * H3_ + col];
    float gz = gi[(size_t)rc * H3_ + H_ + col];
    float gn = gi[(size_t)rc * H3_ + 2 * H_ + col];
    float rr = sigm(gr + ar[r] + br);
    float zz = sigm(gz + az[r] + bz);
    float nn = tanhf(gn + rr * (an[r] + bn));
    float hv = h0old[(size_t)rc * H_ + col];
    h0new[(size_t)rc * H_ + col] = (1.0f - zz) * nn + zz * hv;
  }
}

// GRU layer 1: gi1 = h0' @ W_ih1^T + b_ih1; gh1 = h1 @ W_hh1^T + b_hh1 -> h1'
__global__ void k_gates1(const float* __restrict__ h0new,
                         const float* __restrict__ W_ih1, const float* __restrict__ b_ih1,
                         const float* __restrict__ W_hh1, const float* __restrict__ b_hh1,
                         const float* __restrict__ h1old, float* __restrict__ h1new) {
  int lane = threadIdx.x & 31;
  int rb = threadIdx.x >> 5;
  int cb = blockIdx.x;
  int m = lane & 15, hi = lane >> 4;
  const float* aI = h0new + (size_t)(rb * 16 + m) * H_ + 2 * hi;
  const float* aH = h1old + (size_t)(rb * 16 + m) * H_ + 2 * hi;
  int col = cb * 16 + m;
  v8f ir = {}, iz = {}, in_ = {}, hr = {}, hz = {}, hn = {};
  ir  = wmma_f32_k(aI, W_ih1 + (size_t)(0 * H_ + col) * H_ + 2 * hi, H_, ir);
  iz  = wmma_f32_k(aI, W_ih1 + (size_t)(1 * H_ + col) * H_ + 2 * hi, H_, iz);
  in_ = wmma_f32_k(aI, W_ih1 + (size_t)(2 * H_ + col) * H_ + 2 * hi, H_, in_);
  hr  = wmma_f32_k(aH, W_hh1 + (size_t)(0 * H_ + col) * H_ + 2 * hi, H_, hr);
  hz  = wmma_f32_k(aH, W_hh1 + (size_t)(1 * H_ + col) * H_ + 2 * hi, H_, hz);
  hn  = wmma_f32_k(aH, W_hh1 + (size_t)(2 * H_ + col) * H_ + 2 * hi, H_, hn);
  float bir = b_ih1[col], biz = b_ih1[H_ + col], bin_ = b_ih1[2 * H_ + col];
  float bhr = b_hh1[col], bhz = b_hh1[H_ + col], bhn = b_hh1[2 * H_ + col];
#pragma unroll
  for (int r = 0; r < 8; ++r) {
    int rc = rb * 16 + r + 8 * hi;
    float rr = sigm(ir[r] + bir + hr[r] + bhr);
    float zz = sigm(iz[r] + biz + hz[r] + bhz);
    float nn = tanhf(in_[r] + bin_ + rr * (hn[r] + bhn));
    float hv = h1old[(size_t)rc * H_ + col];
    h1new[(size_t)rc * H_ + col] = (1.0f - zz) * nn + zz * hv;
  }
}

// Luong 'general' attention with PREVIOUS h1: q = h1_old@Wa^T; softmax scores; ctx
__global__ void k_attn(const float* __restrict__ h1old, const float* __restrict__ attn_wa,
                       const float* __restrict__ enc_out, float* __restrict__ ctx) {
  __shared__ float q[H_];
  __shared__ float aw[S_];
  int b = blockIdx.x, tid = threadIdx.x;
  const float* hb = h1old + (size_t)b * H_;
  for (int j = tid; j < H_; j += 256) {
    float s = 0.f;
    const float* w = attn_wa + (size_t)j * H_;
    for (int k = 0; k < H_; ++k) s += hb[k] * w[k];
    q[j] = s;
  }
  __syncthreads();
  const float* eb = enc_out + (size_t)b * S_ * H_;
  if (tid < S_) {
    float s = 0.f;
    const float* e = eb + (size_t)tid * H_;
    for (int k = 0; k < H_; ++k) s += e[k] * q[k];
    aw[tid] = s;
  }
  __syncthreads();
  if (tid == 0) {
    float mx = aw[0];
    for (int s = 1; s < S_; ++s) mx = fmaxf(mx, aw[s]);
    float sum = 0.f;
    for (int s = 0; s < S_; ++s) { aw[s] = expf(aw[s] - mx); sum += aw[s]; }
    float inv = 1.0f / sum;
    for (int s = 0; s < S_; ++s) aw[s] *= inv;
  }
  __syncthreads();
  for (int j = tid; j < H_; j += 256) {
    float s = 0.f;
    for (int k = 0; k < S_; ++k) s += aw[k] * eb[(size_t)k * H_ + j];
    ctx[(size_t)b * H_ + j] = s;
  }
}

// o[t] = tanh([h1', ctx] @ cat_w^T)  (K=1024 split across the two halves)
__global__ void k_out(int t, const float* __restrict__ h1new, const float* __restrict__ ctx,
                      const float* __restrict__ cat_w, float* __restrict__ o) {
  int lane = threadIdx.x & 31;
  int rb = threadIdx.x >> 5;
  int cb = blockIdx.x;
  int m = lane & 15, hi = lane >> 4;
  int col = cb * 16 + m;
  v8f acc = {};
  acc = wmma_f32_k(h1new + (size_t)(rb * 16 + m) * H_ + 2 * hi,
                   cat_w + (size_t)col * (2 * H_) + 2 * hi, H_, acc);
  acc = wmma_f32_k(ctx + (size_t)(rb * 16 + m) * H_ + 2 * hi,
                   cat_w + (size_t)col * (2 * H_) + H_ + 2 * hi, H_, acc);
#pragma unroll
  for (int r = 0; r < 8; ++r) {
    int rc = rb * 16 + r + 8 * hi;
    o[((size_t)t * B_ + rc) * H_ + col] = tanhf(acc[r]);
  }
}

// ---------------------------------------------------------------------------
// logits = o @ fc_w^T + fc_b  -> d_out in [B,T,V] order.
// Workgroup = 8 waves = 128 rows x 64 cols. The 64-column fc_w panel is staged
// into LDS in 64-wide K-chunks by the Tensor Data Mover (double-buffered,
// TENSORcnt-tracked) and shared by all 8 waves -> 8x less L2 traffic, and the
// B-copy runs asynchronously to the WMMAs.
// ---------------------------------------------------------------------------
#define FCK_ 64                  // K-chunk
#define NCHUNK_ (H_ / FCK_)      // 8

#ifdef HAVE_TDM
__device__ __forceinline__ void fc_tdm_issue(const float* fc_w, int cb, int kc,
                                             unsigned lds_byte) {
  // D# group 0: count=1 (user), lds_addr, global_addr (byte), type=2 (image)
  unsigned long long ga =
      (unsigned long long)(uintptr_t)(fc_w + (size_t)(cb * 64) * H_ + kc);
  u32x4 g0 = {1u, lds_byte, (unsigned)(ga & 0xFFFFFFFFu),
              (unsigned)((ga >> 32) & 0x01FFFFFFu) | (2u << 30)};
  // D# group 1: data_size=4B(2); tensor_dim0=512, tensor_dim1=32000;
  // tile_dim0=64, tile_dim1=64, tile_dim2=0; tensor_dim0_stride=512
  i32x8 g1 = {(int)(2u << 16),
              (int)((512u & 0xFFFFu) << 16),
              (int)(((unsigned)32000u & 0xFFFFu) << 16),
              (int)(((unsigned)32000u >> 16) | (64u << 16)),
              (int)64,
              (int)512,
              0, 0};
  i32x4 g2 = {0, 0, 0, 0};   // 2-D tile: dims 2..4 unused
  i32x4 g3 = {0, 0, 0, 0};
  i32x8 g4 = {0, 0, 0, 0, 0, 0, 0, 0};   // unused trailing group (6-arg form)
  __builtin_amdgcn_tensor_load_to_lds(g0, g1, g2, g3, g4, 0);
}
#endif

__global__ void k_fc(const float* __restrict__ o, const float* __restrict__ fc_w,
                     const float* __restrict__ fc_b, float* __restrict__ out) {
  __shared__ float ldsB[2][64 * FCK_];          // 2 x 16KB
  int lane = threadIdx.x & 31;
  int wv = threadIdx.x >> 5;                    // 8 waves
  int rb = blockIdx.x / 500;                    // 16 row-groups of 128
  int cb = blockIdx.x % 500;                    // 500 col-panels of 64
  int m = lane & 15, hi = lane >> 4;
  int row = rb * 128 + wv * 16 + m;             // row = t*B + b
  const float* aPtr = o + (size_t)row * H_ + 2 * hi;
  v8f acc[4] = {};

#ifdef HAVE_TDM
  bool issuer = (threadIdx.x < 32);
  if (issuer) fc_tdm_issue(fc_w, cb, 0, (unsigned)(uintptr_t)&ldsB[0][0]);
#endif
  for (int ic = 0; ic < NCHUNK_; ++ic) {
    int p = ic & 1;
#ifdef HAVE_TDM
    if (issuer) {
      if (ic + 1 < NCHUNK_) {
        fc_tdm_issue(fc_w, cb, (ic + 1) * FCK_,
                     (unsigned)(uintptr_t)&ldsB[1 - p][0]);
        __builtin_amdgcn_s_wait_tensorcnt(1);   // chunk ic resident
      } else {
        __builtin_amdgcn_s_wait_tensorcnt(0);   // last chunk resident
      }
    }
    __syncthreads();
#else
    p = 0;
    __syncthreads();
    // cooperative fallback copy: 64 rows x 64 floats
    for (int i = threadIdx.x; i < 64 * (FCK_ / 4); i += 256) {
      int rrow = i / (FCK_ / 4), kk = (i % (FCK_ / 4)) * 4;
      const float4 vsrc =
          *(const float4*)(fc_w + (size_t)(cb * 64 + rrow) * H_ + ic * FCK_ + kk);
      *(float4*)&ldsB[0][rrow * FCK_ + kk] = vsrc;
    }
    __syncthreads();
#endif
    const float* a0 = aPtr + ic * FCK_;
    const float* bp0 = &ldsB[p][(0 * 16 + m) * FCK_ + 2 * hi];
    const float* bp1 = &ldsB[p][(1 * 16 + m) * FCK_ + 2 * hi];
    const float* bp2 = &ldsB[p][(2 * 16 + m) * FCK_ + 2 * hi];
    const float* bp3 = &ldsB[p][(3 * 16 + m) * FCK_ + 2 * hi];
#pragma unroll
    for (int ks = 0; ks < FCK_; ks += 4) {
      v2f av = *(const v2f*)(a0 + ks);
      acc[0] = __builtin_amdgcn_wmma_f32_16x16x4_f32(false, av, false,
                 *(const v2f*)(bp0 + ks), (short)0, acc[0], false, false);
      acc[1] = __builtin_amdgcn_wmma_f32_16x16x4_f32(false, av, false,
                 *(const v2f*)(bp1 + ks), (short)0, acc[1], false, false);
      acc[2] = __builtin_amdgcn_wmma_f32_16x16x4_f32(false, av, false,
                 *(const v2f*)(bp2 + ks), (short)0, acc[2], false, false);
      acc[3] = __builtin_amdgcn_wmma_f32_16x16x4_f32(false, av, false,
                 *(const v2f*)(bp3 + ks), (short)0, acc[3], false, false);
    }
    __syncthreads();   // protect LDS buffer before it is refilled
  }

#pragma unroll
  for (int r = 0; r < 8; ++r) {
    int rc = rb * 128 + wv * 16 + r + 8 * hi;   // rc = t*B + b
    int tt = rc >> 5, bb = rc & 31;
    size_t orow = (size_t)bb * T_ + tt;         // [B,T,V] layout
#pragma unroll
    for (int c = 0; c < 4; ++c) {
      int col = cb * 64 + c * 16 + m;
      out[orow * V_ + col] = acc[c][r] + fc_b[col];
    }
  }
}

// In-place log_softmax over V per (b,t) row.
__global__ void k_lsm(float* __restrict__ out) {
  __shared__ float red[256];
  int tid = threadIdx.x;
  float* p = out + (size_t)blockIdx.x * V_;
  float mx = -3.402823466e38f;
  for (int i = tid; i < V_; i += 256) mx = fmaxf(mx, p[i]);
  red[tid] = mx; __syncthreads();
  for (int s = 128; s > 0; s >>= 1) {
    if (tid < s) red[tid] = fmaxf(red[tid], red[tid + s]);
    __syncthreads();
  }
  mx = red[0]; __syncthreads();
  float sum = 0.f;
  for (int i = tid; i < V_; i += 256) sum += expf(p[i] - mx);
  red[tid] = sum; __syncthreads();
  for (int s = 128; s > 0; s >>= 1) {
    if (tid < s) red[tid] += red[tid + s];
    __syncthreads();
  }
  float lse = mx + logf(red[0]);
  __syncthreads();
  for (int i = tid; i < V_; i += 256) p[i] -= lse;
}

__global__ void k_hfinal(const float* __restrict__ h0, const float* __restrict__ h1,
                         float* __restrict__ out) {
  int i = blockIdx.x * 256 + threadIdx.x;
  if (i < B_ * H_) {
    out[(size_t)B_ * T_ * V_ + i] = h0[i];
    out[(size_t)B_ * T_ * V_ + B_ * H_ + i] = h1[i];
  }
}

extern "C" void kernel_launch(void* const* d_in, const int* in_sizes, int n_in,
                              void* d_out, int out_size, void* d_ws, size_t ws_size,
                              hipStream_t stream) {
  const int*   target  = (const int*)d_in[0];
  const float* enc_h   = (const float*)d_in[1];
  const float* enc_o   = (const float*)d_in[2];
  const float* emb     = (const float*)d_in[3];
  const float* W_ih0   = (const float*)d_in[4];
  const float* W_hh0   = (const float*)d_in[5];
  const float* b_ih0   = (const float*)d_in[6];
  const float* b_hh0   = (const float*)d_in[7];
  const float* W_ih1   = (const float*)d_in[8];
  const float* W_hh1   = (const float*)d_in[9];
  const float* b_ih1   = (const float*)d_in[10];
  const float* b_hh1   = (const float*)d_in[11];
  const float* attn_wa = (const float*)d_in[12];
  const float* cat_w   = (const float*)d_in[13];
  const float* fc_w    = (const float*)d_in[14];
  const float* fc_b    = (const float*)d_in[15];
  float* out = (float*)d_out;

  // workspace layout (floats): gi0 | o | h0 x2 | h1 x2 | ctx  (~17.1 MB)
  float* gi0 = (float*)d_ws;
  float* o   = gi0 + (size_t)ROWS_ * H3_;
  float* h0b = o + (size_t)ROWS_ * H_;
  float* h1b = h0b + 2 * B_ * H_;
  float* ctx = h1b + 2 * B_ * H_;

  k_init<<<(B_ * H_ + 255) / 256, 256, 0, stream>>>(enc_h, h0b, h1b);
  k_pre<<<1536, 256, 0, stream>>>(target, emb, W_ih0, b_ih0, gi0);

  for (int t = 0; t < T_; ++t) {
    float* h0old = h0b + (t & 1) * B_ * H_;
    float* h0new = h0b + ((t + 1) & 1) * B_ * H_;
    float* h1old = h1b + (t & 1) * B_ * H_;
    float* h1new = h1b + ((t + 1) & 1) * B_ * H_;
    k_gates0<<<32, 64, 0, stream>>>(t, gi0, W_hh0, b_hh0, h0old, h0new);
    k_gates1<<<32, 64, 0, stream>>>(h0new, W_ih1, b_ih1, W_hh1, b_hh1, h1old, h1new);
    k_attn<<<32, 256, 0, stream>>>(h1old, attn_wa, enc_o, ctx);
    k_out<<<32, 64, 0, stream>>>(t, h1new, ctx, cat_w, o);
  }

  k_fc<<<16 * 500, 256, 0, stream>>>(o, fc_w, fc_b, out);
  k_lsm<<<ROWS_, 256, 0, stream>>>(out);
  k_hfinal<<<(B_ * H_ + 255) / 256, 256, 0, stream>>>(h0b, h1b, out);
}